// USGC_7232724927275
// MI455X (gfx1250) — compile-verified
//
#include <hip/hip_runtime.h>
#include <hip/hip_bf16.h>
#include <stdint.h>

typedef __attribute__((ext_vector_type(2))) float v2f;
typedef __attribute__((ext_vector_type(8))) float v8f;

#define FEAT 128
#define NCLS 64

// ---------------------------------------------------------------------------
// fill a u32 buffer with a constant (used for deg=1 init and zeroing tmps)
// ---------------------------------------------------------------------------
__global__ void gcn_fill_u32(unsigned int* __restrict__ p, unsigned int v, long long n) {
    long long i = (long long)blockIdx.x * blockDim.x + threadIdx.x;
    long long stride = (long long)gridDim.x * blockDim.x;
    for (; i < n; i += stride) p[i] = v;
}

// ---------------------------------------------------------------------------
// degree count over destination (col) indices; deg pre-initialized to 1
// (self-loop).  edge_index is int64 [2, E]; col = second row.
// ---------------------------------------------------------------------------
__global__ void gcn_degree(const long long* __restrict__ ei,
                           unsigned int* __restrict__ deg,
                           long long E) {
    long long i = (long long)blockIdx.x * blockDim.x + threadIdx.x;
    if (i >= E) return;
    int c = (int)ei[E + i];
    atomicAdd(&deg[c], 1u);
}

// dinv[i] = rsqrt(deg[i])   (deg >= 1 always, due to self-loop) — in place
__global__ void gcn_dinv(unsigned int* __restrict__ degf, long long n) {
    long long i = (long long)blockIdx.x * blockDim.x + threadIdx.x;
    if (i >= n) return;
    unsigned int d = degf[i];
    float r = __frsqrt_rn((float)d);
    ((float*)degf)[i] = r;
}

// ---------------------------------------------------------------------------
// One wave32 per (edge | self-loop).  Lane l handles features [4l, 4l+4):
// a single float4 load of src[row] and 4 float atomic adds into dst[col].
// work id < E : edge (row=ei[w], col=ei[E+w], norm = dinv[r]*dinv[c])
// work id >= E: self-loop node i=w-E (norm = dinv[i]^2)
// ---------------------------------------------------------------------------
__global__ void gcn_spmm(const float* __restrict__ src,
                         float* __restrict__ dst,
                         const long long* __restrict__ ei,
                         const float* __restrict__ dinv,
                         long long E, long long N) {
    int lane = threadIdx.x & 31;
    long long w = (long long)blockIdx.x * (blockDim.x >> 5) + (threadIdx.x >> 5);
    if (w >= E + N) return;

    int r, c;
    float norm;
    if (w < E) {
        r = (int)ei[w];
        c = (int)ei[E + w];
        norm = dinv[r] * dinv[c];
    } else {
        r = c = (int)(w - E);
        float d = dinv[r];
        norm = d * d;
    }

    const float4* sp = (const float4*)(src + (size_t)r * FEAT);
    float4 v = sp[lane];                       // global_load_b128, 512B/wave
    float* dp = dst + (size_t)c * FEAT + lane * 4;
    atomicAdd(dp + 0, norm * v.x);
    atomicAdd(dp + 1, norm * v.y);
    atomicAdd(dp + 2, norm * v.z);
    atomicAdd(dp + 3, norm * v.w);
}

// ---------------------------------------------------------------------------
// out[M,64] = X[M,128] @ W[64,128]^T + b      (fp32 WMMA 16x16x4)
// One wave per 16x16 output tile; M = 100000 = 6250*16, so EXEC stays all-1s.
// A 16x4 f32 frag: lane&15 = M row; (lane>>4)*2 = K sub-offset (v0,v1 = K,K+1)
// B  4x16 f32 frag: lane&15 = N col;  same K striping (B[k][n] = W[n][k])
// C/D: vgpr j -> M = j + 8*(lane>=16), N = lane&15
// ---------------------------------------------------------------------------
__global__ void gcn_gemm_wmma(const float* __restrict__ X,
                              const float* __restrict__ Wt,   // [64,128] row-major
                              const float* __restrict__ bias, // [64]
                              float* __restrict__ out,        // [M,64]
                              long long M) {
    int lane = threadIdx.x & 31;
    long long tile = (long long)blockIdx.x * (blockDim.x >> 5) + (threadIdx.x >> 5);
    long long ntiles = (M / 16) * (NCLS / 16);
    if (tile >= ntiles) return;

    long long mt = tile >> 2;          // NCLS/16 == 4 tiles in N
    int nt = (int)(tile & 3);
    long long m0 = mt * 16;
    int n0 = nt * 16;

    int lrow = lane & 15;              // M row for A frag, N col for B frag
    int kk = (lane >> 4) * 2;          // K sub-offset within the x4 step

    const float* aRow = X + (m0 + lrow) * (long long)FEAT + kk;
    const float* bRow = Wt + (size_t)(n0 + lrow) * FEAT + kk;

    v8f acc = {};
#pragma unroll
    for (int k = 0; k < FEAT; k += 4) {
        v2f a, b;
        a.x = aRow[k];
        a.y = aRow[k + 1];
        b.x = bRow[k];
        b.y = bRow[k + 1];
        acc = __builtin_amdgcn_wmma_f32_16x16x4_f32(
            /*neg_a=*/false, a, /*neg_b=*/false, b,
            /*c_mod=*/(short)0, acc, /*reuse_a=*/false, /*reuse_b=*/false);
    }

    int nc = n0 + (lane & 15);
    long long mbase = m0 + ((lane >> 4) << 3);
    float bv = bias[nc];
#pragma unroll
    for (int j = 0; j < 8; ++j) {
        out[(mbase + j) * (long long)NCLS + nc] = acc[j] + bv;
    }
}

// ---------------------------------------------------------------------------
extern "C" void kernel_launch(void* const* d_in, const int* in_sizes, int n_in,
                              void* d_out, int out_size, void* d_ws, size_t ws_size,
                              hipStream_t stream) {
    const float* x       = (const float*)d_in[0];
    const long long* ei  = (const long long*)d_in[1];
    const float* W       = (const float*)d_in[2];
    const float* bias    = (const float*)d_in[3];
    float* out           = (float*)d_out;

    const long long N = in_sizes[0] / FEAT;     // 100000
    const long long E = in_sizes[1] / 2;        // 1600000

    // scratch layout: [ dinv/deg : N u32 ] [ tmp1 : N*128 f32 ] [ tmp2 : N*128 f32 ]
    char* ws = (char*)d_ws;
    size_t off_dinv = 0;
    size_t off_tmp1 = ((size_t)N * 4 + 255) & ~(size_t)255;
    size_t off_tmp2 = off_tmp1 + (size_t)N * FEAT * 4;
    unsigned int* deg = (unsigned int*)(ws + off_dinv);
    float* dinv       = (float*)(ws + off_dinv);
    float* tmp1       = (float*)(ws + off_tmp1);
    float* tmp2       = (float*)(ws + off_tmp2);

    const int BT = 256;

    // 1) deg = 1 (self-loops), tmp1 = tmp2 = 0
    {
        long long n = N;
        gcn_fill_u32<<<(unsigned)((n + BT - 1) / BT), BT, 0, stream>>>(deg, 1u, n);
        long long nf = N * FEAT;
        unsigned grid = (unsigned)((nf + (long long)BT * 8 - 1) / ((long long)BT * 8));
        gcn_fill_u32<<<grid, BT, 0, stream>>>((unsigned int*)tmp1, 0u, nf);
        gcn_fill_u32<<<grid, BT, 0, stream>>>((unsigned int*)tmp2, 0u, nf);
    }

    // 2) degree over col, then dinv = rsqrt(deg)
    gcn_degree<<<(unsigned)((E + BT - 1) / BT), BT, 0, stream>>>(ei, deg, E);
    gcn_dinv<<<(unsigned)((N + BT - 1) / BT), BT, 0, stream>>>(deg, N);

    // 3) two hops of normalized SpMM (one wave32 per edge/self-loop)
    {
        long long waves = E + N;
        unsigned grid = (unsigned)((waves + (BT / 32) - 1) / (BT / 32));
        gcn_spmm<<<grid, BT, 0, stream>>>(x, tmp1, ei, dinv, E, N);
        gcn_spmm<<<grid, BT, 0, stream>>>(tmp1, tmp2, ei, dinv, E, N);
    }

    // 4) final linear via fp32 WMMA: out = tmp2 @ W^T + b
    {
        long long ntiles = (N / 16) * (NCLS / 16);   // 25000
        unsigned grid = (unsigned)((ntiles + (BT / 32) - 1) / (BT / 32));
        gcn_gemm_wmma<<<grid, BT, 0, stream>>>(tmp2, W, bias, out, N);
    }
}